// QuantumLayer_19834158973527
// MI455X (gfx1250) — compile-verified
//
#include <hip/hip_runtime.h>
#include <math.h>

#define NQ 4
#define DIM 16
#define QDEPTH 6

typedef float v8f __attribute__((ext_vector_type(8)));
typedef float v2f __attribute__((ext_vector_type(2)));

#if defined(__has_builtin)
#if __has_builtin(__builtin_amdgcn_tanhf)
#define FAST_TANH(x) __builtin_amdgcn_tanhf(x)
#endif
#endif
#ifndef FAST_TANH
#define FAST_TANH(x) tanhf(x)
#endif

// ---- tiny basis-vector circuit simulator (builds fixed 16x16 matrix M) ----
__device__ __forceinline__ void cnot16(float* st, int cm, int tm) {
#pragma unroll
  for (int i = 0; i < DIM; ++i) {
    if ((i & cm) && !(i & tm)) {
      float t = st[i];
      st[i] = st[i | tm];
      st[i | tm] = t;
    }
  }
}

__device__ __forceinline__ void ry16(float* st, int m, float c, float s) {
#pragma unroll
  for (int i = 0; i < DIM; ++i) {
    if (!(i & m)) {
      int j = i | m;
      float s0 = st[i], s1 = st[j];
      st[i] = c * s0 - s * s1;  // mat = [[c,-s],[s,c]]
      st[j] = s * s0 + c * s1;
    }
  }
}

__launch_bounds__(256)
__global__ void quantum_layer_kernel(const float* __restrict__ x,
                                     const float* __restrict__ qp,
                                     const float* __restrict__ scale,
                                     float* __restrict__ out,
                                     int numTiles) {
  __shared__ float Ms[DIM * DIM];  // Ms[j*16+k] = M[j][k], u = M v

  // Build M: column t = circuit applied to basis state e_t.
  // Wire w <-> bit (3-w): wire0->mask8, wire1->4, wire2->2, wire3->1.
  if (threadIdx.x < DIM) {
    float st[DIM];
#pragma unroll
    for (int i = 0; i < DIM; ++i) st[i] = 0.0f;
    st[threadIdx.x] = 1.0f;
    for (int k = 0; k < QDEPTH; ++k) {
      cnot16(st, 8, 4);  // CNOT(0,1)
      cnot16(st, 2, 1);  // CNOT(2,3)
      cnot16(st, 4, 2);  // CNOT(1,2)
#pragma unroll
      for (int w = 0; w < NQ; ++w) {
        float th = 0.5f * qp[k * NQ + w];
        ry16(st, 8 >> w, cosf(th), sinf(th));
      }
    }
#pragma unroll
    for (int j = 0; j < DIM; ++j) Ms[j * DIM + threadIdx.x] = st[j];
  }
  __syncthreads();

  const int lane = threadIdx.x & 31;
  const int n = lane & 15;   // sample within tile / column N
  const int h = lane >> 4;   // half-wave id

  // A-matrix fragments (ISA 16x4 f32 layout): lane<16: row=lane, K={4c,4c+1};
  // lane>=16: row=lane-16, K={4c+2,4c+3}.
  v2f A[4];
#pragma unroll
  for (int c = 0; c < 4; ++c) {
    int off = n * DIM + c * 4 + h * 2;  // 8-byte aligned
    A[c] = *(const v2f*)&Ms[off];
  }

  const float sc = scale[0];
  const int wavesPerBlock = blockDim.x >> 5;
  const int waveId = blockIdx.x * wavesPerBlock + (threadIdx.x >> 5);
  const int nWaves = gridDim.x * wavesPerBlock;

  const float4* __restrict__ x4 = (const float4*)x;
  float4* __restrict__ o4 = (float4*)out;
  const float PI4 = 0.78539816339744831f;

  for (int tile = waveId; tile < numTiles; tile += nWaves) {
    const int s = tile * 16 + n;
    float4 xv = x4[s];

    // angles: a_w = tanh(scale*x_w/||x||)*pi/4 + pi/4
    float inv = rsqrtf(xv.x * xv.x + xv.y * xv.y + xv.z * xv.z + xv.w * xv.w);
    float m = sc * inv;
    float a0 = FAST_TANH(xv.x * m) * PI4 + PI4;
    float a1 = FAST_TANH(xv.y * m) * PI4 + PI4;
    float a2 = FAST_TANH(xv.z * m) * PI4 + PI4;
    float a3 = FAST_TANH(xv.w * m) * PI4 + PI4;
    float c0 = __cosf(a0), s0v = __sinf(a0);
    float c1 = __cosf(a1), s1v = __sinf(a1);
    float c2 = __cosf(a2), s2v = __sinf(a2);
    float c3 = __cosf(a3), s3v = __sinf(a3);

    // B fragment for chunk c holds v_{4c+2h}, v_{4c+2h+1}:
    // i = b0*8+b1*4+b2*2+b3 with b0=c>>1, b1=c&1, b2=h.
    float f2h = h ? s2v : c2;
    float g0 = f2h * c3, g1 = f2h * s3v;
    float p01[4] = {c0 * c1, c0 * s1v, s0v * c1, s0v * s1v};

    v8f acc = {0.f, 0.f, 0.f, 0.f, 0.f, 0.f, 0.f, 0.f};
#pragma unroll
    for (int c = 0; c < 4; ++c) {
      v2f b;
      b[0] = p01[c] * g0;
      b[1] = p01[c] * g1;
      // D[j][sample] += A(16x4) x B(4x16)
      acc = __builtin_amdgcn_wmma_f32_16x16x4_f32(
          false, A[c], false, b, (short)0, acc, false, false);
    }

    // probs: lane holds states j = g + 8h (g = 0..7) of sample n
    float p0 = acc[0] * acc[0], p1 = acc[1] * acc[1];
    float p2 = acc[2] * acc[2], p3 = acc[3] * acc[3];
    float p4 = acc[4] * acc[4], p5 = acc[5] * acc[5];
    float p6 = acc[6] * acc[6], p7 = acc[7] * acc[7];

    // signed Walsh partials: out[p] = sum_j probs_j * (1 - 2*bit_{3-p}(j))
    float a01 = p0 + p1, a23 = p2 + p3, a45 = p4 + p5, a67 = p6 + p7;
    float b01 = p0 - p1, b23 = p2 - p3, b45 = p4 - p5, b67 = p6 - p7;
    float L = a01 + a23, H = a45 + a67;
    float S = L + H;
    float P0 = h ? -S : S;                    // bit3(j) = h
    float P1 = L - H;                         // bit2
    float P2 = (a01 - a23) + (a45 - a67);     // bit1
    float P3 = (b01 + b23) + (b45 + b67);     // bit0

    float r0 = P0 + __shfl_xor(P0, 16, 32);
    float r1 = P1 + __shfl_xor(P1, 16, 32);
    float r2 = P2 + __shfl_xor(P2, 16, 32);
    float r3 = P3 + __shfl_xor(P3, 16, 32);

    if (h == 0) {
      o4[s] = make_float4(r0, r1, r2, r3);
    }
  }
}

extern "C" void kernel_launch(void* const* d_in, const int* in_sizes, int n_in,
                              void* d_out, int out_size, void* d_ws, size_t ws_size,
                              hipStream_t stream) {
  const float* x = (const float*)d_in[0];      // [B,4]
  const float* qp = (const float*)d_in[1];     // [24]
  const float* scale = (const float*)d_in[2];  // [1]
  float* out = (float*)d_out;                  // [B,4]

  const int B = in_sizes[0] / 4;
  const int numTiles = B / 16;  // B = 2^21 -> 131072 tiles

  int blocks = 2048;  // 16384 waves, 8 tiles/wave at B=2M
  int blocksNeeded = (numTiles + 7) / 8;
  if (blocks > blocksNeeded) blocks = blocksNeeded;
  if (blocks < 1) blocks = 1;

  quantum_layer_kernel<<<blocks, 256, 0, stream>>>(x, qp, scale, out, numTiles);
}